// AncProbsLayer_57200374448411
// MI455X (gfx1250) — compile-verified
//
#include <hip/hip_runtime.h>
#include <hip/hip_bf16.h>

// Sizes from the reference
#define B_  512
#define L_  1024
#define K_  8
#define NR_ 512
#define S_  20
#define EXT_ 26

typedef __attribute__((ext_vector_type(2))) float v2f;
typedef __attribute__((ext_vector_type(4))) float v4f;
typedef __attribute__((ext_vector_type(8))) float v8f;

#define LDSTRIDE 33   // 32x32 tiles padded to stride 33 to avoid LDS bank conflicts

__device__ __forceinline__ float softplus_f(float x) {
    return fmaxf(x, 0.0f) + log1pf(expf(-fabsf(x)));
}

// ---------------------------------------------------------------------------
// Kernel 1: build Q[k] (padded to 32x32, zero elsewhere) and softplus(tau).
// blocks 0..7  -> Q for k = blockIdx
// blocks 8..9  -> tau table
// ---------------------------------------------------------------------------
__global__ __launch_bounds__(256) void setup_kernel(
    const float* __restrict__ exch, const float* __restrict__ freq,
    const float* __restrict__ tauk, float* __restrict__ Qws,
    float* __restrict__ tauws) {
    const int blk = blockIdx.x;
    const int t = threadIdx.x;

    if (blk >= K_) {
        int idx = (blk - K_) * 256 + t;
        if (idx < NR_) tauws[idx] = softplus_f(tauk[idx]);
        return;
    }

    __shared__ float sQt[S_ * S_];
    __shared__ float sdiag[S_];
    __shared__ float sinv;

    const float* E = exch + blk * (S_ * S_);
    for (int idx = t; idx < S_ * S_; idx += 256) {
        int i = idx / S_, j = idx % S_;
        float r = 0.0f;
        if (i != j) r = softplus_f(0.5f * (E[i * S_ + j] + E[j * S_ + i]));
        sQt[idx] = r * freq[j];             // Q = R * frequencies (col-wise freq scale)
    }
    __syncthreads();
    if (t < S_) {
        float d = 0.0f;
        for (int j = 0; j < S_; ++j) d += sQt[t * S_ + j];
        sdiag[t] = d;                        // row sums
    }
    __syncthreads();
    if (t == 0) {
        float mue = 0.0f;
        for (int i = 0; i < S_; ++i) mue += freq[i] * sdiag[i];
        sinv = 1.0f / fmaxf(mue, 1e-16f);
    }
    __syncthreads();
    // write padded 32x32 (zeros outside 20x20; diagonal = -rowsum)
    for (int idx = t; idx < 32 * 32; idx += 256) {
        int i = idx >> 5, j = idx & 31;
        float v = 0.0f;
        if (i < S_ && j < S_)
            v = ((i == j) ? -sdiag[i] : sQt[i * S_ + j]) * sinv;
        Qws[blk * 1024 + idx] = v;
    }
}

// ---------------------------------------------------------------------------
// Wave-level 32x32 f32 matmul Z = X * Y using V_WMMA_F32_16X16X4_F32.
// Matrices live in LDS with row stride LDSTRIDE. One wave32, EXEC all ones.
//
// A-fragment (16x4, 2 VGPRs):  lanes 0-15 hold (M=lane, K=kc*4+{0,1}),
//                              lanes 16-31 hold (M=lane-16, K=kc*4+{2,3})
// B-fragment (4x16, 2 VGPRs):  lanes 0-15 hold (K=kc*4+{0,1}, N=lane),
//                              lanes 16-31 hold (K=kc*4+{2,3}, N=lane-16)
// D/C (16x16, 8 VGPRs): VGPR r: lanes 0-15 -> (M=r, N=lane),
//                               lanes 16-31 -> (M=r+8, N=lane-16)
// ---------------------------------------------------------------------------
__device__ __forceinline__ void wave_matmul32(float* Z, const float* X,
                                              const float* Y, int lane) {
    const int lid = lane & 15;
    const int half = lane >> 4;
#pragma unroll
    for (int ti = 0; ti < 2; ++ti) {
#pragma unroll
        for (int tj = 0; tj < 2; ++tj) {
            v8f acc = {};
            const int arow = (ti * 16 + lid) * LDSTRIDE;
#pragma unroll
            for (int kc = 0; kc < 8; ++kc) {
                const int ka = kc * 4 + half * 2;
                v2f a, b;
                a.x = X[arow + ka];
                a.y = X[arow + ka + 1];
                b.x = Y[ka * LDSTRIDE + tj * 16 + lid];
                b.y = Y[(ka + 1) * LDSTRIDE + tj * 16 + lid];
                acc = __builtin_amdgcn_wmma_f32_16x16x4_f32(
                    false, a, false, b, (short)0, acc, false, false);
            }
            const int dbase = (ti * 16 + half * 8) * LDSTRIDE + tj * 16 + lid;
#pragma unroll
            for (int r = 0; r < 8; ++r) Z[dbase + r * LDSTRIDE] = acc[r];
        }
    }
}

// ---------------------------------------------------------------------------
// Kernel 2: one wave per (b,k): P = expm(tau[b] * Q[k]) via scaling-and-
// squaring (order-16 Taylor, 6 squarings). Stores compact 20x20 result.
// ---------------------------------------------------------------------------
__global__ __launch_bounds__(32) void expm_kernel(
    const int* __restrict__ rate_idx, const float* __restrict__ tauws,
    const float* __restrict__ Qws, float* __restrict__ Pws) {
    const int bk = blockIdx.x;          // bk = b*K + k
    const int b = bk >> 3;
    const int k = bk & 7;
    const int lane = threadIdx.x;

    __shared__ float sA[32 * LDSTRIDE];
    __shared__ float sT[32 * LDSTRIDE];
    __shared__ float sP[32 * LDSTRIDE];
    __shared__ float sTmp[32 * LDSTRIDE];

    const float tau = tauws[rate_idx[b]];
    const float scale = tau * (1.0f / 64.0f);   // 2^-SQ, SQ = 6
    const float* Qk = Qws + k * 1024;

    for (int idx = lane; idx < 1024; idx += 32) {
        const int i = idx >> 5, j = idx & 31;
        const float a = Qk[idx] * scale;
        sA[i * LDSTRIDE + j] = a;
        sT[i * LDSTRIDE + j] = a;
        sP[i * LDSTRIDE + j] = a + ((i == j) ? 1.0f : 0.0f);  // I + A (32x32 eye: exact)
    }
    __syncthreads();

    // Taylor: T = T*A/i ; P += T   for i = 2..16
    for (int it = 2; it <= 16; ++it) {
        wave_matmul32(sTmp, sT, sA, lane);
        const float inv = 1.0f / (float)it;
        for (int idx = lane; idx < 1024; idx += 32) {
            const int i = idx >> 5, j = idx & 31;
            const float tv = sTmp[i * LDSTRIDE + j] * inv;
            sT[i * LDSTRIDE + j] = tv;
            sP[i * LDSTRIDE + j] += tv;
        }
        __syncthreads();
    }
    // 6 squarings
    for (int sq = 0; sq < 6; ++sq) {
        wave_matmul32(sTmp, sP, sP, lane);
        for (int idx = lane; idx < 1024; idx += 32) {
            const int i = idx >> 5, j = idx & 31;
            sP[i * LDSTRIDE + j] = sTmp[i * LDSTRIDE + j];
        }
        __syncthreads();
    }
    // write compact 20x20 block
    float* Pout = Pws + (size_t)bk * (S_ * S_);
    for (int idx = lane; idx < S_ * S_; idx += 32) {
        const int i = idx / S_, j = idx % S_;
        Pout[idx] = sP[i * LDSTRIDE + j];
    }
}

// ---------------------------------------------------------------------------
// Kernel 3: streaming writer. out[b,l,k,s] (s in [0,26)) =
//   inputs[b,l] < 20 : (s < 20 ? P[b,k,inputs,s] : 0)
//   else             : (s == inputs[b,l] ? 1 : 0)
// One 16B NT-store per thread; a 4-float group never crosses a (b,l) row
// (row = 208 floats, 208 % 4 == 0).
// ---------------------------------------------------------------------------
__global__ __launch_bounds__(256) void write_kernel(
    const int* __restrict__ inputs, const float* __restrict__ Pws,
    float* __restrict__ out, unsigned total4) {
    const unsigned i4 = blockIdx.x * 256u + threadIdx.x;
    if (i4 >= total4) return;
    const unsigned idx0 = i4 * 4u;
    const unsigned bl = idx0 / (K_ * EXT_);      // = b*L + l
    const unsigned b = bl >> 10;                 // L = 1024
    const int std = inputs[bl];

    v4f v;
#pragma unroll
    for (int e = 0; e < 4; ++e) {
        const unsigned idx = idx0 + e;
        const unsigned r = idx / EXT_;
        const unsigned s = idx - r * EXT_;
        const unsigned k = r & (K_ - 1);
        float val;
        if (std < S_) {
            val = (s < S_)
                ? Pws[(((b << 3) + k) * S_ + (unsigned)std) * S_ + s]
                : 0.0f;
        } else {
            val = (s == (unsigned)std) ? 1.0f : 0.0f;
        }
        v[e] = val;
    }
    __builtin_nontemporal_store(v, (v4f*)(out + idx0));
}

// ---------------------------------------------------------------------------
extern "C" void kernel_launch(void* const* d_in, const int* in_sizes, int n_in,
                              void* d_out, int out_size, void* d_ws, size_t ws_size,
                              hipStream_t stream) {
    const int* inputs = (const int*)d_in[0];        // (B,L) int32
    const int* rate_indices = (const int*)d_in[1];  // (B,) int32
    const float* tau_kernel = (const float*)d_in[2];// (NR,)
    const float* exch = (const float*)d_in[3];      // (K,S,S)
    const float* freq = (const float*)d_in[4];      // (S,)
    float* out = (float*)d_out;

    // workspace layout (floats): Q (8*1024) | tau (512) | P (4096*400) ~ 6.6 MB
    float* Qws = (float*)d_ws;
    float* tauws = Qws + K_ * 1024;
    float* Pws = tauws + NR_;

    setup_kernel<<<K_ + (NR_ + 255) / 256, 256, 0, stream>>>(exch, freq,
                                                             tau_kernel, Qws,
                                                             tauws);
    expm_kernel<<<B_ * K_, 32, 0, stream>>>(rate_indices, tauws, Qws, Pws);

    const unsigned total4 = (unsigned)out_size / 4u;  // out_size = 109,051,904
    write_kernel<<<(total4 + 255) / 256, 256, 0, stream>>>(inputs, Pws, out,
                                                           total4);
}